// GPCwSTU_72576357368005
// MI455X (gfx1250) — compile-verified
//
#include <hip/hip_runtime.h>

// ---------------- problem dims (fixed by the reference) ----------------
#define T_DIM   2048
#define N_DIM   1024
#define MC_DIM  512
#define KF_DIM  20
#define M_HIST  5

// sequential-stage blocking
#define BS       32
#define NBLK     (T_DIM / BS)
#define SEQ_WGS  64
#define SEQ_THR  1024

typedef float v2f __attribute__((ext_vector_type(2)));
typedef float v8f __attribute__((ext_vector_type(8)));

#if defined(__gfx1250__) && __has_builtin(__builtin_amdgcn_wmma_f32_16x16x4_f32)
#define HAVE_WMMA_F32X4 1
#else
#define HAVE_WMMA_F32X4 0
#endif

// D = A(16x4, f32) * B(4x16, f32) + C  -- one wave, fp32-exact
__device__ __forceinline__ v8f wmma4(v2f a, v2f b, v8f c)
{
#if HAVE_WMMA_F32X4
    // 8 args: (neg_a, A, neg_b, B, c_mod, C, reuse_a, reuse_b)
    return __builtin_amdgcn_wmma_f32_16x16x4_f32(false, a, false, b, (short)0, c,
                                                 false, false);
#else
    // shuffle-emulated fallback (dead on gfx1250; keeps other passes compiling)
    const int lane = threadIdx.x & 31;
    const int hi   = lane >> 4;
    #pragma unroll
    for (int k = 0; k < 4; ++k) {
        const int srcHalf = (k & 2) << 3;     // K>=2 lives in lanes 16..31
        float bv = __shfl((k & 1) ? b.y : b.x, (lane & 15) + srcHalf, 32);
        #pragma unroll
        for (int r = 0; r < 8; ++r) {
            float av = __shfl((k & 1) ? a.y : a.x, (r + 8 * hi) + srcHalf, 32);
            c[r] += av * bv;
        }
    }
    return c;
#endif
}

// ---------------------------------------------------------------------------
// Stage a 16x16 f32 tile of row-major B (ldb floats per row) into LDS with the
// CDNA5 async LDS-DMA path. Per wave: 2 x (32 lanes x 16B) = exactly 1KB.
// LDS layout: row-major [k][n], 64B rows.
// ---------------------------------------------------------------------------
__device__ __forceinline__ void async_stage_tile16(const float* __restrict__ gsrc,
                                                   int ldb, unsigned ldsBase,
                                                   int lane)
{
    const int r   = lane >> 2;   // 0..7  (tile row)
    const int seg = lane & 3;    // 0..3  (16B segment within the 64B row)
    const unsigned long long g0 =
        (unsigned long long)(uintptr_t)(gsrc + (size_t)r * ldb + seg * 4);
    const unsigned long long g1 =
        (unsigned long long)(uintptr_t)(gsrc + (size_t)(r + 8) * ldb + seg * 4);
    const unsigned l0 = ldsBase + (unsigned)((r * 16 + seg * 4) * 4);
    const unsigned l1 = l0 + 8 * 16 * 4;
#if defined(__gfx1250__)
    asm volatile("global_load_async_to_lds_b128 %0, %1, off"
                 :: "v"(l0), "v"(g0) : "memory");
    asm volatile("global_load_async_to_lds_b128 %0, %1, off"
                 :: "v"(l1), "v"(g1) : "memory");
    asm volatile("s_wait_asynccnt 0x0" ::: "memory");
#else
    (void)g0; (void)g1; (void)l0; (void)l1;
#endif
}

// ---------------------------------------------------------------------------
// Kernel 1: P[t,c] = bias[c] + sum_{i,n} w[t-4+i, n] * E[c,n,i]
// GEMM view: M=T, N=MC, K = M_HIST*N_DIM = 5120, k = i*N + n (shifted-window A)
// Boundary rows handled with clamped loads + v_cndmask (no EXEC divergence).
// ---------------------------------------------------------------------------
__global__ void __launch_bounds__(128)
compute_P_kernel(const float* __restrict__ w, const float* __restrict__ E,
                 const float* __restrict__ bias, float* __restrict__ P)
{
    const int lane = threadIdx.x;
    const int tile = blockIdx.x * blockDim.y + threadIdx.y;
    const int tilesN = MC_DIM / 16;
    const int tm = tile / tilesN, tn = tile % tilesN;
    const int lo = lane & 15, hi = lane >> 4;
    const int t = tm * 16 + lo;          // A row for this lane
    const int c = tn * 16 + lo;          // B/C col for this lane
    v8f acc = {};
    for (int k0 = 0; k0 < M_HIST * N_DIM; k0 += 4) {
        const int kk = k0 + 2 * hi;          // pair base: n, n+1 contiguous
        const int i  = kk >> 10;             // N_DIM = 1024
        const int n  = kk & (N_DIM - 1);
        const int rw = t - (M_HIST - 1) + i;
        const int rc = rw >= 0 ? rw : 0;     // clamped, always-valid address
        v2f a = *(const v2f*)(w + (size_t)rc * N_DIM + n);
        if (rw < 0) { a.x = 0.0f; a.y = 0.0f; }
        v2f b;
        b.x = E[c * (N_DIM * M_HIST) + n * M_HIST + i];
        b.y = E[c * (N_DIM * M_HIST) + (n + 1) * M_HIST + i];
        acc = wmma4(a, b, acc);
    }
    const float bc = bias[c];
    #pragma unroll
    for (int r = 0; r < 8; ++r)
        P[(tm * 16 + r + 8 * hi) * MC_DIM + c] = acc[r] + bc;
}

// ---------------------------------------------------------------------------
// Kernel 2 (batched over z=i): H_i[c,c'] = sum_n Estu[i,c,n] * K[c',n]
// C = A * B^T; both operands k-contiguous -> pure global_load_b64 traffic.
// ---------------------------------------------------------------------------
__global__ void __launch_bounds__(128)
gemm_rt_f32(const float* __restrict__ A, const float* __restrict__ B,
            float* __restrict__ C, int Mdim, int Ndim, int Kdim,
            long strideA, long strideC)
{
    const float* Ab = A + (long)blockIdx.z * strideA;
    float*       Cb = C + (long)blockIdx.z * strideC;
    const int lane = threadIdx.x;
    const int tile = blockIdx.x * blockDim.y + threadIdx.y;
    const int tilesN = Ndim >> 4;
    const int tm = tile / tilesN, tn = tile % tilesN;
    if (tm * 16 >= Mdim) return;         // wave-uniform
    const int lo = lane & 15, hi = lane >> 4;
    const int row = tm * 16 + lo;
    const int col = tn * 16 + lo;
    v8f acc = {};
    for (int k0 = 0; k0 < Kdim; k0 += 4) {
        const int ka = k0 + 2 * hi;
        v2f a = *(const v2f*)(Ab + (size_t)row * Kdim + ka);
        v2f b = *(const v2f*)(B  + (size_t)col * Kdim + ka);  // B^T[k,col]=B[col,k]
        acc = wmma4(a, b, acc);
    }
    #pragma unroll
    for (int r = 0; r < 8; ++r)
        Cb[(long)(tm * 16 + r + 8 * hi) * Ndim + col] = acc[r];
}

// ---------------------------------------------------------------------------
// Generic C = A(row-major MxK) * B(row-major KxN): used for X*Q and U*R.
// B tiles staged to LDS via async LDS-DMA; 4 WMMAs per 16-k chunk.
// ---------------------------------------------------------------------------
__global__ void __launch_bounds__(128)
gemm_rr_f32(const float* __restrict__ A, const float* __restrict__ B,
            float* __restrict__ C, int Mdim, int Ndim, int Kdim)
{
    __shared__ alignas(16) float Bt[4][16][16];
    const int lane = threadIdx.x;
    const int tile = blockIdx.x * blockDim.y + threadIdx.y;
    const int tilesN = Ndim >> 4;
    const int tm = tile / tilesN, tn = tile % tilesN;
    if (tm * 16 >= Mdim) return;
    const int lo = lane & 15, hi = lane >> 4;
    const int row = tm * 16 + lo;
    const int n0  = tn * 16;
    float* tileB = &Bt[threadIdx.y][0][0];
    const unsigned ldsBase = (unsigned)(uintptr_t)tileB;
    v8f acc = {};
    for (int k0 = 0; k0 < Kdim; k0 += 16) {
        async_stage_tile16(B + (size_t)k0 * Ndim + n0, Ndim, ldsBase, lane);
        #pragma unroll
        for (int m = 0; m < 4; ++m) {
            const int ka = 4 * m + 2 * hi;
            v2f a = *(const v2f*)(A + (size_t)row * Kdim + k0 + ka);
            v2f b;
            b.x = tileB[ka * 16 + lo];
            b.y = tileB[(ka + 1) * 16 + lo];
            acc = wmma4(a, b, acc);
        }
    }
    #pragma unroll
    for (int r = 0; r < 8; ++r)
        C[(long)(tm * 16 + r + 8 * hi) * Ndim + (n0 + lo)] = acc[r];
}

// ---------------------------------------------------------------------------
// Rm[s,n] = sum_{k=(i,c)} phi[s,i]*U[s,c] * Estu[k*N + n];  K = KF*MC = 10240.
// A generated on the fly (phi scalar x contiguous U pair); B staged via async.
// ---------------------------------------------------------------------------
__global__ void __launch_bounds__(128)
compute_R_kernel(const float* __restrict__ phi, const float* __restrict__ U,
                 const float* __restrict__ Estu, float* __restrict__ Rm)
{
    __shared__ alignas(16) float Bt[4][16][16];
    const int lane = threadIdx.x;
    const int tile = blockIdx.x * blockDim.y + threadIdx.y;
    const int tilesN = N_DIM / 16;
    const int tm = tile / tilesN, tn = tile % tilesN;
    const int lo = lane & 15, hi = lane >> 4;
    const int s  = tm * 16 + lo;
    const int n0 = tn * 16;
    float* tileB = &Bt[threadIdx.y][0][0];
    const unsigned ldsBase = (unsigned)(uintptr_t)tileB;
    v8f acc = {};
    for (int k0 = 0; k0 < KF_DIM * MC_DIM; k0 += 16) {
        async_stage_tile16(Estu + (size_t)k0 * N_DIM + n0, N_DIM, ldsBase, lane);
        const int   i     = k0 >> 9;                  // constant over the chunk
        const float ph    = phi[s * KF_DIM + i];
        const int   cbase = (k0 & (MC_DIM - 1)) + 2 * hi;
        #pragma unroll
        for (int m = 0; m < 4; ++m) {
            v2f a = *(const v2f*)(U + (size_t)s * MC_DIM + cbase + 4 * m);
            a.x *= ph; a.y *= ph;
            const int ka = 4 * m + 2 * hi;
            v2f b;
            b.x = tileB[ka * 16 + lo];
            b.y = tileB[(ka + 1) * 16 + lo];
            acc = wmma4(a, b, acc);
        }
    }
    #pragma unroll
    for (int r = 0; r < 8; ++r)
        Rm[(long)(tm * 16 + r + 8 * hi) * N_DIM + (n0 + lo)] = acc[r];
}

// ---------------------------------------------------------------------------
// Persistent sequential stage: y_{t+1} = y_t + M_t u_t,  u_t = P_t - y_t
// ---------------------------------------------------------------------------
__device__ __forceinline__ void grid_barrier(unsigned* cnt, unsigned* gen,
                                             unsigned nwg)
{
    __syncthreads();
    if (threadIdx.x == 0) {
        const unsigned g =
            __hip_atomic_load(gen, __ATOMIC_ACQUIRE, __HIP_MEMORY_SCOPE_AGENT);
        const unsigned a = __hip_atomic_fetch_add(cnt, 1u, __ATOMIC_ACQ_REL,
                                                  __HIP_MEMORY_SCOPE_AGENT);
        if (a == nwg - 1) {
            __hip_atomic_store(cnt, 0u, __ATOMIC_RELAXED,
                               __HIP_MEMORY_SCOPE_AGENT);
            __hip_atomic_fetch_add(gen, 1u, __ATOMIC_RELEASE,
                                   __HIP_MEMORY_SCOPE_AGENT);
        } else {
            while (__hip_atomic_load(gen, __ATOMIC_ACQUIRE,
                                     __HIP_MEMORY_SCOPE_AGENT) == g)
                __builtin_amdgcn_s_sleep(4);
        }
    }
    __syncthreads();
}

__global__ void __launch_bounds__(SEQ_THR)
stageA_kernel(const float* __restrict__ phi, const float* __restrict__ P,
              const float* __restrict__ H, float* __restrict__ Mblk,
              float* __restrict__ U, unsigned* sync)
{
    __shared__ float y_sh[MC_DIM];
    __shared__ float u_sh[MC_DIM];
    const unsigned tid  = threadIdx.x;
    const unsigned wg   = blockIdx.x;
    const unsigned gsz  = gridDim.x * blockDim.x;
    const unsigned gtid = wg * blockDim.x + tid;
    if (wg == 0)
        for (int c = tid; c < MC_DIM; c += blockDim.x) y_sh[c] = 0.0f;
    __syncthreads();
    unsigned* cnt = sync;
    unsigned* gen = sync + 1;

    for (int blk = 0; blk < NBLK; ++blk) {
        const int t0 = blk * BS;
        // ---- phase a: Mblk[s][c][c'] = sum_i phi[t0+s,i] * H[i][c][c'] ----
        const int total = BS * MC_DIM * MC_DIM;
        for (int e = gtid; e < total; e += gsz) {
            const int cp  = e & (MC_DIM - 1);
            const int rem = e >> 9;
            const int c   = rem & (MC_DIM - 1);
            const int s   = rem >> 9;
            const float* ph = phi + (t0 + s) * KF_DIM;
            float acc = 0.0f;
            #pragma unroll
            for (int i = 0; i < KF_DIM; ++i)
                acc += ph[i] * H[((long)i * MC_DIM + c) * MC_DIM + cp];
            Mblk[e] = acc;
        }
        grid_barrier(cnt, gen, gridDim.x);
        // ---- phase b: WG0 runs the BS sequential steps (y,u in LDS) ----
        if (wg == 0) {
            for (int s = 0; s < BS; ++s) {
                const int t = t0 + s;
                for (int c = tid; c < MC_DIM; c += blockDim.x) {
                    const float u = P[t * MC_DIM + c] - y_sh[c];
                    u_sh[c] = u;
                    U[t * MC_DIM + c] = u;
                }
                __syncthreads();
                if (tid < MC_DIM) {
                    const float* Mrow = Mblk + (long)s * MC_DIM * MC_DIM;
                    float d = 0.0f;
                    for (int c = 0; c < MC_DIM; ++c) {
                        if ((c & 63) == 0 && c + 64 < MC_DIM)
                            __builtin_prefetch(&Mrow[(c + 64) * MC_DIM + tid], 0, 1);
                        d += Mrow[c * MC_DIM + tid] * u_sh[c];
                    }
                    y_sh[tid] += d;
                }
                __syncthreads();
            }
        }
        grid_barrier(cnt, gen, gridDim.x);
    }
}

// ---------------------------------------------------------------------------
// X = exclusive prefix sum of Rm along t (one thread per column)
// ---------------------------------------------------------------------------
__global__ void prefix_X_kernel(const float* __restrict__ Rm, float* __restrict__ X)
{
    const int n = blockIdx.x * blockDim.x + threadIdx.x;
    if (n >= N_DIM) return;
    float acc = 0.0f;
    for (int t = 0; t < T_DIM; ++t) {
        X[(long)t * N_DIM + n] = acc;
        acc += Rm[(long)t * N_DIM + n];
    }
}

// ---------------------------------------------------------------------------
// losses[t] = dot(X[t], (XQ)[t]) + dot(U[t], (UR)[t])
// ---------------------------------------------------------------------------
__global__ void __launch_bounds__(256)
loss_kernel(const float* __restrict__ X, const float* __restrict__ Y,
            const float* __restrict__ U, const float* __restrict__ Z,
            float* __restrict__ out)
{
    const int t = blockIdx.x;
    float s = 0.0f;
    for (int n = threadIdx.x; n < N_DIM; n += blockDim.x)
        s += X[(long)t * N_DIM + n] * Y[(long)t * N_DIM + n];
    for (int c = threadIdx.x; c < MC_DIM; c += blockDim.x)
        s += U[(long)t * MC_DIM + c] * Z[(long)t * MC_DIM + c];
    __shared__ float red[256];
    red[threadIdx.x] = s;
    __syncthreads();
    for (int off = 128; off > 0; off >>= 1) {
        if (threadIdx.x < (unsigned)off) red[threadIdx.x] += red[threadIdx.x + off];
        __syncthreads();
    }
    if (threadIdx.x == 0) out[t] = red[0];
}

__global__ void init_sync_kernel(unsigned* sync)
{
    if (threadIdx.x == 0 && blockIdx.x == 0) { sync[0] = 0u; sync[1] = 0u; }
}

// ---------------------------------------------------------------------------
extern "C" void kernel_launch(void* const* d_in, const int* in_sizes, int n_in,
                              void* d_out, int out_size, void* d_ws, size_t ws_size,
                              hipStream_t stream)
{
    (void)in_sizes; (void)n_in; (void)out_size;
    // setup_inputs order: A, B, Q, R, K, E, bias, E_stu, phi, w_test
    const float* Qm   = (const float*)d_in[2];
    const float* Rc   = (const float*)d_in[3];
    const float* Kg   = (const float*)d_in[4];
    const float* E    = (const float*)d_in[5];
    const float* bias = (const float*)d_in[6];
    const float* Estu = (const float*)d_in[7];
    const float* phi  = (const float*)d_in[8];
    const float* w    = (const float*)d_in[9];
    float* losses = (float*)d_out;

    char* ws = (char*)d_ws;
    size_t off = 0;
    auto alloc = [&](size_t bytes) -> void* {
        void* p = ws + off;
        off += (bytes + 255) & ~(size_t)255;
        return p;
    };
    unsigned* sync = (unsigned*)alloc(256);
    float* P    = (float*)alloc((size_t)T_DIM * MC_DIM * 4);            //  4 MB
    float* H    = (float*)alloc((size_t)KF_DIM * MC_DIM * MC_DIM * 4);  // 21 MB
    float* U    = (float*)alloc((size_t)T_DIM * MC_DIM * 4);            //  4 MB
    float* Mblk = (float*)alloc((size_t)BS * MC_DIM * MC_DIM * 4);      // 34 MB
    float* Rm   = (float*)alloc((size_t)T_DIM * N_DIM * 4);             //  8 MB
    float* X    = (float*)alloc((size_t)T_DIM * N_DIM * 4);             //  8 MB
    float* Y    = (float*)alloc((size_t)T_DIM * N_DIM * 4);             //  8 MB
    float* Z    = (float*)alloc((size_t)T_DIM * MC_DIM * 4);            //  4 MB
    if (off > ws_size) return;  // workspace too small -> no-op

    const dim3 wblk(32, 4, 1);  // 4 waves per block, one 16x16 tile per wave

    init_sync_kernel<<<1, 64, 0, stream>>>(sync);

    // P[t,c] : (T/16)*(MC/16) = 4096 tiles
    compute_P_kernel<<<dim3(4096 / 4), wblk, 0, stream>>>(w, E, bias, P);

    // H_i = Estu_i * K^T : 20 batched 512x512x1024 GEMMs, 1024 tiles each
    gemm_rt_f32<<<dim3(1024 / 4, 1, KF_DIM), wblk, 0, stream>>>(
        Estu, Kg, H, MC_DIM, MC_DIM, N_DIM,
        (long)MC_DIM * N_DIM, (long)MC_DIM * MC_DIM);

    // sequential recurrence (persistent, software grid barrier)
    stageA_kernel<<<dim3(SEQ_WGS), dim3(SEQ_THR), 0, stream>>>(
        phi, P, H, Mblk, U, sync);

    // per-step state increments: 2048x1024 out, K = 10240 -> 8192 tiles
    compute_R_kernel<<<dim3(8192 / 4), wblk, 0, stream>>>(phi, U, Estu, Rm);

    // X = exclusive prefix over t
    prefix_X_kernel<<<dim3(4), dim3(256), 0, stream>>>(Rm, X);

    // Y = X*Q (2048x1024x1024), Z = U*R (2048x512x512)
    gemm_rr_f32<<<dim3(8192 / 4), wblk, 0, stream>>>(X, Qm, Y, T_DIM, N_DIM, N_DIM);
    gemm_rr_f32<<<dim3(4096 / 4), wblk, 0, stream>>>(U, Rc, Z, T_DIM, MC_DIM, MC_DIM);

    // losses[t] = x^T Q x + u^T R u
    loss_kernel<<<dim3(T_DIM), dim3(256), 0, stream>>>(X, Y, U, Z, losses);
}